// MessageGenerator_74758200754699
// MI455X (gfx1250) — compile-verified
//
#include <hip/hip_runtime.h>
#include <hip/hip_bf16.h>

typedef float v2f __attribute__((ext_vector_type(2)));
typedef float v8f __attribute__((ext_vector_type(8)));

// ---------------------------------------------------------------------------
// 0) zero scratch (msg accumulator + row_sum + m) — must run every call
// ---------------------------------------------------------------------------
__global__ void zero_f32(float* __restrict__ p, size_t n) {
    size_t i = (size_t)blockIdx.x * blockDim.x + threadIdx.x;
    size_t stride = (size_t)gridDim.x * blockDim.x;
    for (; i < n; i += stride) p[i] = 0.0f;
}

// ---------------------------------------------------------------------------
// 1) m = max(max(gate), 0)  (non-negative floats -> int-bit atomicMax is exact)
// ---------------------------------------------------------------------------
__global__ void reduce_max_gate(const float* __restrict__ g, int E, float* __restrict__ m) {
    float local = 0.0f;
    for (int i = blockIdx.x * blockDim.x + threadIdx.x; i < E;
         i += gridDim.x * blockDim.x)
        local = fmaxf(local, g[i]);
    #pragma unroll
    for (int off = 16; off > 0; off >>= 1)
        local = fmaxf(local, __shfl_down(local, off, 32));
    if ((threadIdx.x & 31) == 0)
        atomicMax(reinterpret_cast<int*>(m), __float_as_int(local));
}

// ---------------------------------------------------------------------------
// 2) exp_e = exp(gate - m); row_sum[src] += exp_e  (f32 global atomics)
// ---------------------------------------------------------------------------
__global__ void edge_exp(const float* __restrict__ g, const int* __restrict__ idx,
                         const float* __restrict__ m, float* __restrict__ exp_e,
                         float* __restrict__ row_sum, int E) {
    int e = blockIdx.x * blockDim.x + threadIdx.x;
    if (e >= E) return;
    float v = expf(g[e] - m[0]);
    exp_e[e] = v;
    atomicAdd(&row_sum[idx[2 * e]], v);
}

// ---------------------------------------------------------------------------
// WMMA f32 GEMM: C[M,Ncol] = A[M,K] @ B[K,Ncol] + bias, one 16x16 tile/wave.
// A frag (16x4 f32): lanes 0-15 -> K=k,k+1 ; lanes 16-31 -> K=k+2,k+3 (same rows)
// B frag (4x16 f32): mirrored; C/D: VGPR r -> M=r (lanes 0-15), M=r+8 (16-31).
// MODE 0: +bias.  MODE 1: +bias, relu, zero rows with row_sum<=0.
// ---------------------------------------------------------------------------
template <int MODE>
__global__ __launch_bounds__(256) void gemm_wmma_f32(
    const float* __restrict__ A, const float* __restrict__ B,
    const float* __restrict__ bias, float* __restrict__ C,
    const float* __restrict__ row_sum, int K, int Ncol, int colTiles, int totalTiles) {
    const int lane = threadIdx.x & 31;
    const int wave = threadIdx.x >> 5;
    const int tile = blockIdx.x * 8 + wave;
    if (tile >= totalTiles) return;            // wave-uniform: EXEC stays all-ones
    const int rowTile = tile / colTiles;
    const int colTile = tile % colTiles;
    const int row0 = rowTile * 16;
    const int col0 = colTile * 16;
    const int half = lane >> 4;                // K sub-pair select
    const int l    = lane & 15;

    const float* __restrict__ Arow = A + (size_t)(row0 + l) * K;
    const float* __restrict__ Bcol = B + col0 + l;

    v8f acc = {};
    for (int k = 0; k < K; k += 4) {
        const int ka = k + half * 2;
        v2f a, b;
        a.x = Arow[ka];
        a.y = Arow[ka + 1];
        b.x = Bcol[(size_t)ka * Ncol];
        b.y = Bcol[(size_t)(ka + 1) * Ncol];
        acc = __builtin_amdgcn_wmma_f32_16x16x4_f32(
            /*neg_a=*/false, a, /*neg_b=*/false, b,
            /*c_mod=*/(short)0, acc, /*reuse_a=*/false, /*reuse_b=*/false);
    }

    const float bv = bias[col0 + l];
    #pragma unroll
    for (int r = 0; r < 8; ++r) {
        const int row = row0 + r + half * 8;
        float v = acc[r] + bv;
        if (MODE == 1) {
            v = fmaxf(v, 0.0f);
            v = (row_sum[row] > 0.0f) ? v : 0.0f;
        }
        C[(size_t)row * Ncol + col0 + l] = v;
    }
}

// ---------------------------------------------------------------------------
// 5) edge scatter: msg[src][0:256]   += atten * Mfeat[dst]
//                  msg[dst][256:512] += atten * Mfeat[src]
//    64 threads per edge, 4 columns (float4) per thread.
// ---------------------------------------------------------------------------
__global__ __launch_bounds__(256) void edge_scatter(
    const int* __restrict__ idx, const float* __restrict__ exp_e,
    const float* __restrict__ row_sum, const float* __restrict__ Mfeat,
    float* __restrict__ msg, int E) {
    const long long gid = (long long)blockIdx.x * blockDim.x + threadIdx.x;
    const int e = (int)(gid >> 6);
    if (e >= E) return;
    const int c = (int)(gid & 63) << 2;
    const int s = idx[2 * e];
    const int d = idx[2 * e + 1];
    const float atten = exp_e[e] / (row_sum[s] + 1e-6f);

    const float4 md = *reinterpret_cast<const float4*>(Mfeat + (size_t)d * 256 + c);
    float* p0 = msg + (size_t)s * 512 + c;
    atomicAdd(p0 + 0, atten * md.x);
    atomicAdd(p0 + 1, atten * md.y);
    atomicAdd(p0 + 2, atten * md.z);
    atomicAdd(p0 + 3, atten * md.w);

    const float4 ms = *reinterpret_cast<const float4*>(Mfeat + (size_t)s * 256 + c);
    float* p1 = msg + (size_t)d * 512 + 256 + c;
    atomicAdd(p1 + 0, atten * ms.x);
    atomicAdd(p1 + 1, atten * ms.y);
    atomicAdd(p1 + 2, atten * ms.z);
    atomicAdd(p1 + 3, atten * ms.w);
}

// ---------------------------------------------------------------------------
// 7) per-row LayerNorm (128 cols) + ReLU, in place. One 128-thread block/row.
// ---------------------------------------------------------------------------
__global__ __launch_bounds__(128) void ln_relu(float* __restrict__ h,
                                               const float* __restrict__ gamma,
                                               const float* __restrict__ beta) {
    __shared__ float red[128];
    const int row = blockIdx.x;
    const int t = threadIdx.x;
    const float x = h[(size_t)row * 128 + t];

    red[t] = x;
    __syncthreads();
    #pragma unroll
    for (int s = 64; s > 0; s >>= 1) {
        if (t < s) red[t] += red[t + s];
        __syncthreads();
    }
    const float mu = red[0] * (1.0f / 128.0f);
    __syncthreads();

    const float dx = x - mu;
    red[t] = dx * dx;
    __syncthreads();
    #pragma unroll
    for (int s = 64; s > 0; s >>= 1) {
        if (t < s) red[t] += red[t + s];
        __syncthreads();
    }
    const float var = red[0] * (1.0f / 128.0f);

    const float y = dx * rsqrtf(var + 1e-5f) * gamma[t] + beta[t];
    h[(size_t)row * 128 + t] = fmaxf(y, 0.0f);
}

// ---------------------------------------------------------------------------
extern "C" void kernel_launch(void* const* d_in, const int* in_sizes, int n_in,
                              void* d_out, int out_size, void* d_ws, size_t ws_size,
                              hipStream_t stream) {
    const float* src_feat = (const float*)d_in[0];   // (N,512)
    const float* gate     = (const float*)d_in[1];   // (E,)
    const int*   idx      = (const int*)  d_in[2];   // (E,2)
    const float* Wm       = (const float*)d_in[3];   // (512,256)
    const float* bm       = (const float*)d_in[4];   // (256,)
    const float* W1       = (const float*)d_in[5];   // (512,128)
    const float* b1       = (const float*)d_in[6];   // (128,)
    const float* gamma    = (const float*)d_in[7];   // (128,)
    const float* beta     = (const float*)d_in[8];   // (128,)
    const float* W2       = (const float*)d_in[9];   // (128,512)
    const float* b2       = (const float*)d_in[10];  // (512,)

    const int N = in_sizes[0] / 512;                 // 20000
    const int E = in_sizes[1];                       // 200000

    // Workspace layout (floats). msg/row_sum/m are contiguous so one zero pass
    // re-initializes all accumulators every call (graph replays don't re-poison).
    float* ws       = (float*)d_ws;
    float* msg      = ws;                                   // N*512
    float* row_sum  = msg + (size_t)N * 512;                // N
    float* m        = row_sum + N;                          // 1 (+7 pad)
    float* Mfeat    = m + 8;                                // N*256
    float* exp_e    = Mfeat + (size_t)N * 256;              // E
    float* h        = exp_e + E;                            // N*128
    const size_t zeroCount = (size_t)N * 512 + N + 8;

    zero_f32<<<2048, 256, 0, stream>>>(msg, zeroCount);
    reduce_max_gate<<<256, 256, 0, stream>>>(gate, E, m);
    edge_exp<<<(E + 255) / 256, 256, 0, stream>>>(gate, idx, m, exp_e, row_sum, E);

    // Mfeat = src_feat @ Wm + bm   (N x 512 x 256)
    {
        const int colTiles = 256 / 16, totalTiles = (N / 16) * colTiles;
        gemm_wmma_f32<0><<<(totalTiles + 7) / 8, 256, 0, stream>>>(
            src_feat, Wm, bm, Mfeat, nullptr, 512, 256, colTiles, totalTiles);
    }

    // edge scatter into msg (concat[out0,out1])
    {
        const long long threads = (long long)E * 64;
        edge_scatter<<<(unsigned)((threads + 255) / 256), 256, 0, stream>>>(
            idx, exp_e, row_sum, Mfeat, msg, E);
    }

    // h = msg @ W1 + b1   (N x 512 x 128)
    {
        const int colTiles = 128 / 16, totalTiles = (N / 16) * colTiles;
        gemm_wmma_f32<0><<<(totalTiles + 7) / 8, 256, 0, stream>>>(
            msg, W1, b1, h, nullptr, 512, 128, colTiles, totalTiles);
    }

    // LayerNorm + ReLU in place on h
    ln_relu<<<N, 128, 0, stream>>>(h, gamma, beta);

    // out = mask(relu(h @ W2 + b2))   (N x 128 x 512)
    {
        const int colTiles = 512 / 16, totalTiles = (N / 16) * colTiles;
        gemm_wmma_f32<1><<<(totalTiles + 7) / 8, 256, 0, stream>>>(
            h, W2, b2, (float*)d_out, row_sum, 128, 512, colTiles, totalTiles);
    }
}